// GraphSAGEModel_80195629350956
// MI455X (gfx1250) — compile-verified
//
#include <hip/hip_runtime.h>
#include <hip/hip_bf16.h>

// GraphSAGE 2-layer on MI455X (gfx1250), wave32, f32 WMMA 16x16x4.
// Aggregation is L2-resident (x:19MB, h:26MB << 192MB L2) -> scatter-add with
// native no-return global_atomic_add_f32; GEMMs use the f32 matrix path.

typedef float v2f __attribute__((ext_vector_type(2)));
typedef float v8f __attribute__((ext_vector_type(8)));

constexpr int NNODES = 50000;   // 3125 * 16, exact M-tiling
constexpr int CIN    = 96;
constexpr int CHID   = 128;
constexpr int COUT   = 64;

// Workspace layout (floats)
constexpr size_t OFF_DEG  = 0;
constexpr size_t OFF_AGG1 = 50048;                       // deg (N) padded
constexpr size_t OFF_H    = OFF_AGG1 + (size_t)NNODES * CIN;
constexpr size_t OFF_P    = OFF_H    + (size_t)NNODES * CHID;
constexpr size_t OFF_AGGP = OFF_P    + (size_t)NNODES * COUT;

// Guaranteed-native f32 atomic add: relaxed, device scope, result unused ->
// lowers to no-return global_atomic_add_f32 (no CAS loop).
__device__ __forceinline__ void atomic_add_f32(float* p, float v) {
    __hip_atomic_fetch_add(p, v, __ATOMIC_RELAXED, __HIP_MEMORY_SCOPE_AGENT);
}

// ---------------- zero two regions (grid-stride) ----------------
__global__ void zero2_kernel(float* p0, long n0, float* p1, long n1) {
    long i = (long)blockIdx.x * blockDim.x + threadIdx.x;
    long stride = (long)gridDim.x * blockDim.x;
    for (long j = i; j < n0; j += stride) p0[j] = 0.0f;
    for (long j = i; j < n1; j += stride) p1[j] = 0.0f;
}

// ---------------- edge scatter-add, 96-wide + degree ----------------
__global__ __launch_bounds__(256) void scatter96_kernel(
    const float* __restrict__ x, const int* __restrict__ ei, long E,
    float* __restrict__ agg, float* __restrict__ deg)
{
    long e = (long)blockIdx.x * 8 + (threadIdx.x >> 5);   // one wave32 per edge
    int lane = threadIdx.x & 31;
    if (e >= E) return;
    int src = ei[e];
    int dst = ei[E + e];
    const float* xs = x + (long)src * CIN;
    float* ad = agg + (long)dst * CIN;
    atomic_add_f32(&ad[lane],      xs[lane]);
    atomic_add_f32(&ad[lane + 32], xs[lane + 32]);
    atomic_add_f32(&ad[lane + 64], xs[lane + 64]);
    if (lane == 0) atomic_add_f32(&deg[dst], 1.0f);
}

// ---------------- edge scatter-add, 64-wide (projected layer 2) ----------------
__global__ __launch_bounds__(256) void scatter64_kernel(
    const float* __restrict__ p, const int* __restrict__ ei, long E,
    float* __restrict__ agg)
{
    long e = (long)blockIdx.x * 8 + (threadIdx.x >> 5);
    int lane = threadIdx.x & 31;
    if (e >= E) return;
    int src = ei[e];
    int dst = ei[E + e];
    const float* ps = p + (long)src * COUT;
    float* ad = agg + (long)dst * COUT;
    atomic_add_f32(&ad[lane],      ps[lane]);
    atomic_add_f32(&ad[lane + 32], ps[lane + 32]);
}

// ---------------- layer 1: h = relu((agg/deg)@W_l + x@W_r + b) ----------------
// Block: 256 threads = 8 waves. M-tile 16, each wave owns a 16-wide N-tile (8*16=128).
__global__ __launch_bounds__(256) void gemm1_kernel(
    const float* __restrict__ x, const float* __restrict__ agg,
    const float* __restrict__ deg,
    const float* __restrict__ wl, const float* __restrict__ wr,
    const float* __restrict__ b1, float* __restrict__ h)
{
    __shared__ float s_a[16][100];   // mean-aggregated rows (pad 100: conflict-free)
    __shared__ float s_x[16][100];
    __shared__ float s_inv[16];

    const int row0 = blockIdx.x * 16;
    const int t = threadIdx.x;
    if (t < 16) s_inv[t] = 1.0f / fmaxf(deg[row0 + t], 1.0f);
    __syncthreads();
    for (int i = t; i < 16 * CIN; i += 256) {
        int r = i / CIN, c = i % CIN;
        s_a[r][c] = agg[(size_t)(row0 + r) * CIN + c] * s_inv[r];
        s_x[r][c] = x[(size_t)(row0 + r) * CIN + c];
    }
    __syncthreads();

    const int lane  = t & 31;
    const int col0  = (t >> 5) * 16;        // wave's N-tile
    const int mrow  = lane & 15;            // A: M index
    const int khalf = (lane >> 4) << 1;     // A/B: K sub-pair (0 or 2)
    const int ncol  = lane & 15;            // B/D: N index

    v8f acc = {};
    for (int k = 0; k < CIN; k += 4) {       // mean-agg @ W_l
        v2f a; a.x = s_a[mrow][k + khalf]; a.y = s_a[mrow][k + khalf + 1];
        v2f b; b.x = wl[(size_t)(k + khalf) * CHID + col0 + ncol];
               b.y = wl[(size_t)(k + khalf + 1) * CHID + col0 + ncol];
        acc = __builtin_amdgcn_wmma_f32_16x16x4_f32(false, a, false, b,
                                                    (short)0, acc, false, false);
    }
    for (int k = 0; k < CIN; k += 4) {       // x @ W_r
        v2f a; a.x = s_x[mrow][k + khalf]; a.y = s_x[mrow][k + khalf + 1];
        v2f b; b.x = wr[(size_t)(k + khalf) * CHID + col0 + ncol];
               b.y = wr[(size_t)(k + khalf + 1) * CHID + col0 + ncol];
        acc = __builtin_amdgcn_wmma_f32_16x16x4_f32(false, a, false, b,
                                                    (short)0, acc, false, false);
    }

    const float bias = b1[col0 + ncol];
    const int rbase = (lane >> 4) * 8;       // D: lanes 16-31 hold M=r+8
    for (int r = 0; r < 8; ++r) {
        float v = fmaxf(acc[r] + bias, 0.0f);
        h[(size_t)(row0 + rbase + r) * CHID + col0 + ncol] = v;
    }
}

// ---------------- p = h @ W2_l  (M tiles of 16, N=64, K=128) ----------------
__global__ __launch_bounds__(128) void gemmp_kernel(
    const float* __restrict__ h, const float* __restrict__ wl,
    float* __restrict__ p)
{
    __shared__ float s_h[16][132];           // pad 132: conflict-free
    const int row0 = blockIdx.x * 16;
    const int t = threadIdx.x;
    for (int i = t; i < 16 * CHID; i += 128) {
        int r = i >> 7, c = i & 127;
        s_h[r][c] = h[(size_t)(row0 + r) * CHID + c];
    }
    __syncthreads();

    const int lane  = t & 31;
    const int col0  = (t >> 5) * 16;         // 4 waves * 16 = 64
    const int mrow  = lane & 15;
    const int khalf = (lane >> 4) << 1;
    const int ncol  = lane & 15;

    v8f acc = {};
    for (int k = 0; k < CHID; k += 4) {
        v2f a; a.x = s_h[mrow][k + khalf]; a.y = s_h[mrow][k + khalf + 1];
        v2f b; b.x = wl[(size_t)(k + khalf) * COUT + col0 + ncol];
               b.y = wl[(size_t)(k + khalf + 1) * COUT + col0 + ncol];
        acc = __builtin_amdgcn_wmma_f32_16x16x4_f32(false, a, false, b,
                                                    (short)0, acc, false, false);
    }
    const int rbase = (lane >> 4) * 8;
    for (int r = 0; r < 8; ++r)
        p[(size_t)(row0 + rbase + r) * COUT + col0 + ncol] = acc[r];
}

// ---------------- out = aggp/deg + h @ W2_r + b2 ----------------
__global__ __launch_bounds__(128) void gemm2_kernel(
    const float* __restrict__ h, const float* __restrict__ aggp,
    const float* __restrict__ deg, const float* __restrict__ wr,
    const float* __restrict__ b2, float* __restrict__ out)
{
    __shared__ float s_h[16][132];
    __shared__ float s_inv[16];
    const int row0 = blockIdx.x * 16;
    const int t = threadIdx.x;
    if (t < 16) s_inv[t] = 1.0f / fmaxf(deg[row0 + t], 1.0f);
    __syncthreads();
    for (int i = t; i < 16 * CHID; i += 128) {
        int r = i >> 7, c = i & 127;
        s_h[r][c] = h[(size_t)(row0 + r) * CHID + c];
    }
    __syncthreads();

    const int lane  = t & 31;
    const int col0  = (t >> 5) * 16;
    const int mrow  = lane & 15;
    const int khalf = (lane >> 4) << 1;
    const int ncol  = lane & 15;

    v8f acc = {};
    for (int k = 0; k < CHID; k += 4) {
        v2f a; a.x = s_h[mrow][k + khalf]; a.y = s_h[mrow][k + khalf + 1];
        v2f b; b.x = wr[(size_t)(k + khalf) * COUT + col0 + ncol];
               b.y = wr[(size_t)(k + khalf + 1) * COUT + col0 + ncol];
        acc = __builtin_amdgcn_wmma_f32_16x16x4_f32(false, a, false, b,
                                                    (short)0, acc, false, false);
    }
    const float bias = b2[col0 + ncol];
    const int rbase = (lane >> 4) * 8;
    for (int r = 0; r < 8; ++r) {
        int m = rbase + r;
        float v = acc[r] + bias
                + aggp[(size_t)(row0 + m) * COUT + col0 + ncol] * s_inv[m];
        out[(size_t)(row0 + m) * COUT + col0 + ncol] = v;
    }
}

extern "C" void kernel_launch(void* const* d_in, const int* in_sizes, int n_in,
                              void* d_out, int out_size, void* d_ws, size_t ws_size,
                              hipStream_t stream) {
    const float* x    = (const float*)d_in[0];
    const int*   ei   = (const int*)d_in[1];
    const float* w1l  = (const float*)d_in[2];
    const float* w1r  = (const float*)d_in[3];
    const float* b1   = (const float*)d_in[4];
    const float* w2l  = (const float*)d_in[5];
    const float* w2r  = (const float*)d_in[6];
    const float* b2   = (const float*)d_in[7];
    float* out = (float*)d_out;
    float* ws  = (float*)d_ws;

    const long E = in_sizes[1] / 2;

    float* deg  = ws + OFF_DEG;
    float* agg1 = ws + OFF_AGG1;
    float* h    = ws + OFF_H;
    float* p    = ws + OFF_P;
    float* aggp = ws + OFF_AGGP;

    // 1) zero deg+agg1 (contiguous) and aggp
    zero2_kernel<<<1024, 256, 0, stream>>>(
        ws, (long)(OFF_AGG1 + (size_t)NNODES * CIN),
        aggp, (long)NNODES * COUT);

    // 2) layer-1 aggregation (96-wide) + degree
    long blocks_e = (E + 7) / 8;
    scatter96_kernel<<<(unsigned)blocks_e, 256, 0, stream>>>(x, ei, E, agg1, deg);

    // 3) layer 1 fused GEMM (mean, dual-K, bias, relu) -> h
    gemm1_kernel<<<NNODES / 16, 256, 0, stream>>>(x, agg1, deg, w1l, w1r, b1, h);

    // 4) project before aggregating (mean is linear): p = h @ W2_l
    gemmp_kernel<<<NNODES / 16, 128, 0, stream>>>(h, w2l, p);

    // 5) layer-2 aggregation on projected 64-wide features
    scatter64_kernel<<<(unsigned)blocks_e, 256, 0, stream>>>(p, ei, E, aggp);

    // 6) out = aggp/deg + h @ W2_r + b2
    gemm2_kernel<<<NNODES / 16, 128, 0, stream>>>(h, aggp, deg, w2r, b2, out);
}